// SliceLoss_87359634801123
// MI455X (gfx1250) — compile-verified
//
#include <hip/hip_runtime.h>

#define G        192
#define WCHUNK   32                 // w rows per block
#define NBLK     (G * (G / WCHUNK)) // 192 * 6 = 1152
#define NTHR     192                // 6 wave32s
#define ITERS    (WCHUNK / 4)       // 8 iterations, 4 rows per iteration

typedef __attribute__((ext_vector_type(2))) float v2f;
typedef __attribute__((ext_vector_type(8))) float v8f;

// ws layout (floats):
//   [   0] sumL_D[192]   [ 192] sumL_M[192]
//   [ 384] sumW_D[192]   [ 576] sumW_M[192]
//   [ 768] sumH_D[192]   [ 960] sumH_M[192]
#define WS_FLOATS 1152

__global__ void zero_ws_kernel(float* __restrict__ ws) {
    int i = blockIdx.x * blockDim.x + threadIdx.x;
    if (i < WS_FLOATS) ws[i] = 0.0f;
}

__global__ __launch_bounds__(NTHR) void slice_reduce_kernel(
    const float* __restrict__ vx,  const float* __restrict__ vy,
    const float* __restrict__ vz,  const float* __restrict__ p,
    const float* __restrict__ mask,
    const float* __restrict__ rvx, const float* __restrict__ rvy,
    const float* __restrict__ rvz, const float* __restrict__ rp,
    float* __restrict__ ws)
{
    __shared__ float wAccD[WCHUNK], wAccM[WCHUNK];
    __shared__ float hAccD[G],      hAccM[G];

    const int t       = threadIdx.x;      // 0..191
    const int b       = blockIdx.x;       // 0..1151
    const int l       = b / (G / WCHUNK); // 0..191
    const int w0      = (b % (G / WCHUNK)) * WCHUNK;
    const int row_sub = t / 48;           // 0..3  (which of 4 rows this iter)
    const int hq      = t % 48;           // float4 index within a row

    if (t < WCHUNK) { wAccD[t] = 0.0f; wAccM[t] = 0.0f; }
    hAccD[t] = 0.0f; hAccM[t] = 0.0f;     // t spans all 192 h bins
    __syncthreads();

    // register accumulators for this thread's 4 fixed h bins
    float hD0 = 0.f, hD1 = 0.f, hD2 = 0.f, hD3 = 0.f;
    float hM0 = 0.f, hM1 = 0.f, hM2 = 0.f, hM3 = 0.f;

    const float4* VX = (const float4*)vx;  const float4* VY = (const float4*)vy;
    const float4* VZ = (const float4*)vz;  const float4* PP = (const float4*)p;
    const float4* MM = (const float4*)mask;
    const float4* RX = (const float4*)rvx; const float4* RY = (const float4*)rvy;
    const float4* RZ = (const float4*)rvz; const float4* RP = (const float4*)rp;

    // float4-granular flat offset: l*192*192/4*... = l*9216 + w*48 + hq
    long off = (long)l * 9216 + (long)(w0 + row_sub) * 48 + hq;

    for (int it = 0; it < ITERS; ++it, off += 4 * 48) {
        float4 a0 = VX[off], a1 = VY[off], a2 = VZ[off], a3 = PP[off];
        float4 r0 = RX[off], r1 = RY[off], r2 = RZ[off], r3 = RP[off];
        float4 fm = MM[off];

        float dx = (r0.x - a0.x) + (r1.x - a1.x) + (r2.x - a2.x) + (r3.x - a3.x);
        float dy = (r0.y - a0.y) + (r1.y - a1.y) + (r2.y - a2.y) + (r3.y - a3.y);
        float dz = (r0.z - a0.z) + (r1.z - a1.z) + (r2.z - a2.z) + (r3.z - a3.z);
        float dw = (r0.w - a0.w) + (r1.w - a1.w) + (r2.w - a2.w) + (r3.w - a3.w);

        hD0 += dx; hD1 += dy; hD2 += dz; hD3 += dw;
        hM0 += fm.x; hM1 += fm.y; hM2 += fm.z; hM3 += fm.w;

        float dS = (dx + dy) + (dz + dw);
        float mS = (fm.x + fm.y) + (fm.z + fm.w);
        // 16-lane segment reduce (each 16-lane group tiles 1/3 of a row)
        #pragma unroll
        for (int s = 1; s < 16; s <<= 1) {
            dS += __shfl_xor(dS, s, 32);
            mS += __shfl_xor(mS, s, 32);
        }
        if ((t & 15) == 0) {
            int wl = 4 * it + row_sub;          // local w bin, 0..31
            atomicAdd(&wAccD[wl], dS);
            atomicAdd(&wAccM[wl], mS);
        }
    }

    // combine per-thread h registers into block-level LDS bins (4-way sharing)
    int hb = 4 * hq;
    atomicAdd(&hAccD[hb + 0], hD0); atomicAdd(&hAccD[hb + 1], hD1);
    atomicAdd(&hAccD[hb + 2], hD2); atomicAdd(&hAccD[hb + 3], hD3);
    atomicAdd(&hAccM[hb + 0], hM0); atomicAdd(&hAccM[hb + 1], hM1);
    atomicAdd(&hAccM[hb + 2], hM2); atomicAdd(&hAccM[hb + 3], hM3);
    __syncthreads();

    float* sumL_D = ws;        float* sumL_M = ws + 192;
    float* sumW_D = ws + 384;  float* sumW_M = ws + 576;
    float* sumH_D = ws + 768;  float* sumH_M = ws + 960;

    atomicAdd(&sumH_D[t], hAccD[t]);
    atomicAdd(&sumH_M[t], hAccM[t]);
    if (t < WCHUNK) {
        atomicAdd(&sumW_D[w0 + t], wAccD[t]);
        atomicAdd(&sumW_M[w0 + t], wAccM[t]);
    }
    if (t == 0) {
        float lD = 0.f, lM = 0.f;
        for (int i = 0; i < WCHUNK; ++i) { lD += wAccD[i]; lM += wAccM[i]; }
        atomicAdd(&sumL_D[l], lD);
        atomicAdd(&sumL_M[l], lM);
    }
}

// One wave32. Per-lane partial ratio sums reduced across lanes with
// v_wmma_f32_16x16x4_f32 against an all-ones B: D[m][n] = acc[m] + acc[m+16].
__global__ __launch_bounds__(32) void finalize_kernel(const float* __restrict__ ws,
                                                      float* __restrict__ out)
{
    const int t = threadIdx.x; // 0..31, full wave, EXEC all ones
    const float* sumL_D = ws;        const float* sumL_M = ws + 192;
    const float* sumW_D = ws + 384;  const float* sumW_M = ws + 576;
    const float* sumH_D = ws + 768;  const float* sumH_M = ws + 960;

    float val = 0.0f;
    #pragma unroll
    for (int k = 0; k < 6; ++k) {
        int i = t + 32 * k;
        val += sumL_D[i] / sumL_M[i];
        val += sumW_D[i] / sumW_M[i];
        val += sumH_D[i] / sumH_M[i];
    }

    // A (16x4 f32): lanes 0-15: VGPR0=K0, VGPR1=K1; lanes 16-31: VGPR0=K2, VGPR1=K3.
    v2f a; a[0] = val;  a[1] = 0.0f;
    v2f bb; bb[0] = 1.0f; bb[1] = 1.0f;     // B = all ones (layout-agnostic)
    v8f c = {};
    v8f d = __builtin_amdgcn_wmma_f32_16x16x4_f32(
        /*neg_a=*/false, a, /*neg_b=*/false, bb,
        /*c_mod=*/(short)0, c, /*reuse_a=*/false, /*reuse_b=*/false);

    // lane 0 holds D[0..7][0], lane 16 holds D[8..15][0]
    float s = ((d[0] + d[1]) + (d[2] + d[3])) + ((d[4] + d[5]) + (d[6] + d[7]));
    float total = __shfl(s, 0, 32) + __shfl(s, 16, 32);
    if (t == 0) out[0] = total * (1.0f / 192.0f);
}

extern "C" void kernel_launch(void* const* d_in, const int* in_sizes, int n_in,
                              void* d_out, int out_size, void* d_ws, size_t ws_size,
                              hipStream_t stream) {
    (void)in_sizes; (void)n_in; (void)out_size; (void)ws_size;
    const float* vx   = (const float*)d_in[0];
    const float* vy   = (const float*)d_in[1];
    const float* vz   = (const float*)d_in[2];
    const float* p    = (const float*)d_in[3];
    const float* mask = (const float*)d_in[4];
    const float* rvx  = (const float*)d_in[5];
    const float* rvy  = (const float*)d_in[6];
    const float* rvz  = (const float*)d_in[7];
    const float* rp   = (const float*)d_in[8];
    float* ws  = (float*)d_ws;
    float* out = (float*)d_out;

    zero_ws_kernel<<<(WS_FLOATS + 255) / 256, 256, 0, stream>>>(ws);
    slice_reduce_kernel<<<NBLK, NTHR, 0, stream>>>(vx, vy, vz, p, mask,
                                                   rvx, rvy, rvz, rp, ws);
    finalize_kernel<<<1, 32, 0, stream>>>(ws, out);
}